// TemporalGCN_76115410419784
// MI455X (gfx1250) — compile-verified
//
#include <hip/hip_runtime.h>

// Problem constants (match reference)
#define T_DIM  12
#define N_DIM  200000
#define E_DIM  3200000
#define KP1    4                 // powers 0..3
#define P_DIM  4
#define TOUT   (T_DIM - P_DIM)   // 8 output time steps

typedef __attribute__((ext_vector_type(2))) float v2f;
typedef __attribute__((ext_vector_type(8))) float v8f;

// px workspace layout: px[(tp*KP1 + k)*N_DIM + n]   (38.4 MB, L2-resident)

// ---------------------------------------------------------------------------
// Kernel 1: initialize px: k=0 plane = x, k=1..3 planes = 0 (atomics need 0s).
// float4 path: b128 loads/stores, T*N/4 = 600k threads.
// ---------------------------------------------------------------------------
__global__ void tgcn_init_px(const float4* __restrict__ x4, float* __restrict__ px) {
  const long total = (long)T_DIM * (N_DIM / 4);
  const float4 z = make_float4(0.f, 0.f, 0.f, 0.f);
  for (long i = (long)blockIdx.x * blockDim.x + threadIdx.x; i < total;
       i += (long)gridDim.x * blockDim.x) {
    int tp = (int)(i / (N_DIM / 4));
    int n4 = (int)(i - (long)tp * (N_DIM / 4));
    float4 v = x4[i];                       // x is [T*N] contiguous
    float4* base = (float4*)px + (long)(tp * KP1) * (N_DIM / 4) + n4;
    base[0]                     = v;        // k = 0
    base[1L * (N_DIM / 4)]      = z;        // k = 1
    base[2L * (N_DIM / 4)]      = z;        // k = 2
    base[3L * (N_DIM / 4)]      = z;        // k = 3
  }
}

// ---------------------------------------------------------------------------
// Kernel 2: one SpMV level across all 12 graphs simultaneously.
//   px[t,k,row] += w * px[t,k-1,col]   via global_atomic_add_f32.
// 4 edges per thread: rows/cols/weights streamed with b128 loads
// (460 MB of edge data per level is the HBM roofline term).
// ---------------------------------------------------------------------------
__global__ void tgcn_spmv_level(const int*   __restrict__ ei,   // [T,2,E]
                                const float* __restrict__ ew,   // [T,E]
                                float* __restrict__ px, int k) {
  const long total = (long)T_DIM * (E_DIM / 4);
  for (long i = (long)blockIdx.x * blockDim.x + threadIdx.x; i < total;
       i += (long)gridDim.x * blockDim.x) {
    int tp = (int)(i / (E_DIM / 4));
    int e4 = (int)(i - (long)tp * (E_DIM / 4));
    const int4*   rp = (const int4*)(ei + (long)tp * 2 * E_DIM) + e4;
    const int4*   cp = (const int4*)(ei + (long)tp * 2 * E_DIM + E_DIM) + e4;
    const float4* wp = (const float4*)(ew + (long)tp * E_DIM) + e4;
    int4   r = *rp;
    int4   c = *cp;
    float4 w = *wp;
    const float* __restrict__ src = px + ((long)tp * KP1 + (k - 1)) * N_DIM;
    float*       __restrict__ dst = px + ((long)tp * KP1 + k)       * N_DIM;
    atomicAdd(&dst[r.x], w.x * src[c.x]);
    atomicAdd(&dst[r.y], w.y * src[c.y]);
    atomicAdd(&dst[r.z], w.z * src[c.z]);
    atomicAdd(&dst[r.w], w.w * src[c.w]);
  }
}

// ---------------------------------------------------------------------------
// Kernel 3: dense combine via V_WMMA_F32_16X16X4_F32.
//   out[n, t] = relu( sum_{tp,k} px[tp,k,n] * H[(tp,k), t] ),
//   H[(tp,k),t] = h[k, tp-t] when 0 <= tp-t < 4 and t < 8, else 0.
// One wave = one 16-node tile; 12 WMMAs (one per tp, K=4 powers each).
// h lives in SGPRs (uniform scalar loads); B entries are built with
// v_cndmask chains in registers -> no LDS, no EXEC toggling between WMMAs.
// ---------------------------------------------------------------------------
__global__ void tgcn_combine_wmma(const float* __restrict__ px,
                                  const float* __restrict__ h,   // [4,4]
                                  float* __restrict__ out) {     // [N, 8]
  int lane = threadIdx.x & 31;
  int wave = threadIdx.x >> 5;
  int tile = blockIdx.x * (blockDim.x >> 5) + wave;   // 16 nodes per tile
  long n0 = (long)tile * 16;

  int hi  = lane >> 4;      // 0: lanes 0-15, 1: lanes 16-31
  int col = lane & 15;      // A: M row; B/D: N column (= output t for col<8)
  int k0  = hi * 2;         // K pair handled by this lane half: {k0, k0+1}
  long node = n0 + col;

  // Uniform h -> SGPRs, then one per-lane select (by lane half) per entry.
  float hx0 = hi ? h[8]  : h[0];   // row k0,   d=0..3
  float hx1 = hi ? h[9]  : h[1];
  float hx2 = hi ? h[10] : h[2];
  float hx3 = hi ? h[11] : h[3];
  float hy0 = hi ? h[12] : h[4];   // row k0+1, d=0..3
  float hy1 = hi ? h[13] : h[5];
  float hy2 = hi ? h[14] : h[6];
  float hy3 = hi ? h[15] : h[7];

  v8f acc = {};
  #pragma unroll
  for (int tp = 0; tp < T_DIM; ++tp) {
    // A tile 16x4: lane half supplies powers k0,k0+1 of node `node` at time tp
    const float* pa = px + ((long)tp * KP1 + k0) * N_DIM + node;
    v2f a; a.x = pa[0]; a.y = pa[N_DIM];
    // B tile 4x16: H[(tp, k0..k0+1), col] via register select chain
    int d = tp - col;
    bool valid = ((unsigned)d < (unsigned)P_DIM) & (col < TOUT);
    float bx = (d == 0) ? hx0 : (d == 1) ? hx1 : (d == 2) ? hx2 : hx3;
    float by = (d == 0) ? hy0 : (d == 1) ? hy1 : (d == 2) ? hy2 : hy3;
    v2f b;
    b.x = valid ? bx : 0.0f;
    b.y = valid ? by : 0.0f;
    // (neg_a, A, neg_b, B, c_mod, C, reuse_a, reuse_b)
    acc = __builtin_amdgcn_wmma_f32_16x16x4_f32(
        false, a, false, b, (short)0, acc, false, false);
  }

  // D layout: VGPR v -> M = v + 8*hi, N = col. Store relu for t = col < 8.
  if (col < TOUT) {
    #pragma unroll
    for (int v = 0; v < 8; ++v) {
      long n = n0 + v + hi * 8;
      float r = acc[v];
      out[n * TOUT + col] = r > 0.0f ? r : 0.0f;
    }
  }
}

// ---------------------------------------------------------------------------
extern "C" void kernel_launch(void* const* d_in, const int* in_sizes, int n_in,
                              void* d_out, int out_size, void* d_ws, size_t ws_size,
                              hipStream_t stream) {
  const float* x  = (const float*)d_in[0];   // [T, N, 1]
  const int*   ei = (const int*)d_in[1];     // [T, 2, E]
  const float* ew = (const float*)d_in[2];   // [T, E]
  const float* h  = (const float*)d_in[3];   // [4, 4]
  float* out = (float*)d_out;                // [N, TOUT]
  float* px  = (float*)d_ws;                 // 12*4*200000 floats = 38.4 MB

  // 1) init px planes (float4)
  {
    long total = (long)T_DIM * (N_DIM / 4);      // 600k
    int blocks = (int)((total + 255) / 256);
    tgcn_init_px<<<blocks, 256, 0, stream>>>((const float4*)x, px);
  }
  // 2) three sequential SpMV levels (each level: 12 independent graphs)
  {
    long total = (long)T_DIM * (E_DIM / 4);      // 9.6M threads, 4 edges each
    int blocks = (int)((total + 255) / 256);
    tgcn_spmv_level<<<blocks, 256, 0, stream>>>(ei, ew, px, 1);
    tgcn_spmv_level<<<blocks, 256, 0, stream>>>(ei, ew, px, 2);
    tgcn_spmv_level<<<blocks, 256, 0, stream>>>(ei, ew, px, 3);
  }
  // 3) WMMA combine: 12500 tiles of 16 nodes, 4 waves (128 thr) per block
  {
    int tiles = N_DIM / 16;                      // 12500, exact
    int blocks = tiles / 4;                      // 3125, exact
    tgcn_combine_wmma<<<blocks, 128, 0, stream>>>(px, h, out);
  }
}